// HybridGraphQCNN_65481071403300
// MI455X (gfx1250) — compile-verified
//
#include <hip/hip_runtime.h>
#include <math.h>

#define N_NODES 100000
#define N_EDGES 3200000
#define NODE_DIM 128
#define HIDDEN 16
#define TILES ((N_NODES + 15) / 16) /* 6250, exact */
#define MLP_WAVES 8
#define MLP_BLOCKS ((TILES + MLP_WAVES - 1) / MLP_WAVES)
#define AGG_BLOCKS 256
#define AGG_THREADS 256

typedef __attribute__((ext_vector_type(2))) float v2f;
typedef __attribute__((ext_vector_type(8))) float v8f;

// ---------------- per-node 3-layer tanh MLP via fp32 WMMA ----------------
__global__ __launch_bounds__(256) void mlp_kernel(
    const float* __restrict__ x,
    const float* __restrict__ W1, const float* __restrict__ b1,
    const float* __restrict__ W2, const float* __restrict__ b2,
    const float* __restrict__ W3, const float* __restrict__ b3,
    float* __restrict__ hout)
{
    __shared__ float W1s[NODE_DIM * HIDDEN];          // 8 KB
    __shared__ float W2s[HIDDEN * HIDDEN];
    __shared__ float W3s[HIDDEN * HIDDEN];
    __shared__ float b1s[HIDDEN], b2s[HIDDEN], b3s[HIDDEN];
    __shared__ __align__(16) float xs[MLP_WAVES][16 * 32];  // 16 KB
    __shared__ float hs[MLP_WAVES][16 * 16];                // 8 KB

    const int tid = threadIdx.x;
    for (int i = tid; i < NODE_DIM * HIDDEN; i += 256) W1s[i] = W1[i];
    if (tid < HIDDEN * HIDDEN) { W2s[tid] = W2[tid]; W3s[tid] = W3[tid]; }
    if (tid < HIDDEN) { b1s[tid] = b1[tid]; b2s[tid] = b2[tid]; b3s[tid] = b3[tid]; }
    __syncthreads();

    const int wave = tid >> 5;
    const int lane = tid & 31;
    const int nidx = lane & 15;          // A: M row; B/C: N column
    const int koff = (lane >> 4) << 1;   // 0 or 2 (K split across half-waves)
    const int hi8  = (lane >> 4) << 3;   // C/D: rows 0..7 or 8..15

    int tile = blockIdx.x * MLP_WAVES + wave;
    const bool store_ok = (tile < TILES);
    if (tile >= TILES) tile = TILES - 1;           // keep all waves uniform
    const int node_base = tile * 16;

    // ---- layer 1: [16x128] @ [128x16] as 32 WMMA f32 16x16x4 steps ----
    v8f c = {};
    for (int kc = 0; kc < NODE_DIM; kc += 32) {
        // stage a 16x32 fp32 chunk of x for this wave (coalesced float4)
        #pragma unroll
        for (int it = 0; it < 4; ++it) {
            int idx = it * 32 + lane;          // 0..127 float4 slots
            int r   = idx >> 3;                // row 0..15
            int c4  = (idx & 7) << 2;          // col 0,4,...,28
            const float4 v =
                *(const float4*)(x + (size_t)(node_base + r) * NODE_DIM + kc + c4);
            *(float4*)(&xs[wave][r * 32 + c4]) = v;
        }
        __syncthreads();
        #pragma unroll
        for (int kk = 0; kk < 32; kk += 4) {
            v2f a, b;
            a.x = xs[wave][nidx * 32 + kk + koff];
            a.y = xs[wave][nidx * 32 + kk + koff + 1];
            const int kg = kc + kk + koff;
            b.x = W1s[(kg + 0) * HIDDEN + nidx];
            b.y = W1s[(kg + 1) * HIDDEN + nidx];
            c = __builtin_amdgcn_wmma_f32_16x16x4_f32(
                    false, a, false, b, (short)0, c, false, false);
        }
        __syncthreads();
    }
    #pragma unroll
    for (int v = 0; v < 8; ++v)
        hs[wave][(v + hi8) * 16 + nidx] = tanhf(c[v] + b1s[nidx]);
    __syncthreads();

    // ---- layers 2 and 3: 16x16 @ 16x16, 4 WMMA steps each ----
    const float* Wsl[2] = { W2s, W3s };
    const float* bsl[2] = { b2s, b3s };
    for (int layer = 0; layer < 2; ++layer) {
        v8f c2 = {};
        #pragma unroll
        for (int kk = 0; kk < 16; kk += 4) {
            v2f a, b;
            a.x = hs[wave][nidx * 16 + kk + koff];
            a.y = hs[wave][nidx * 16 + kk + koff + 1];
            b.x = Wsl[layer][(kk + koff + 0) * 16 + nidx];
            b.y = Wsl[layer][(kk + koff + 1) * 16 + nidx];
            c2 = __builtin_amdgcn_wmma_f32_16x16x4_f32(
                     false, a, false, b, (short)0, c2, false, false);
        }
        __syncthreads();
        #pragma unroll
        for (int v = 0; v < 8; ++v)
            hs[wave][(v + hi8) * 16 + nidx] = tanhf(c2[v] + bsl[layer][nidx]);
        __syncthreads();
    }

    if (store_ok) {
        #pragma unroll
        for (int v = 0; v < 8; ++v)
            hout[(size_t)(node_base + v + hi8) * HIDDEN + nidx] =
                hs[wave][(v + hi8) * 16 + nidx];
    }
}

// ---------------- degree count (integer atomics: deterministic) ----------------
__global__ void zero_deg(unsigned* __restrict__ deg) {
    int i = blockIdx.x * blockDim.x + threadIdx.x;
    if (i < N_NODES) deg[i] = 0u;
}

__global__ void count_deg(const int* __restrict__ src, unsigned* __restrict__ deg) {
    int i = blockIdx.x * blockDim.x + threadIdx.x;
    int stride = gridDim.x * blockDim.x;
    for (int e = i; e < N_EDGES; e += stride)
        atomicAdd(&deg[src[e]], 1u);
}

// ---------------- edge aggregation: per-block fp32 partials (deterministic) ----
__global__ __launch_bounds__(AGG_THREADS) void agg_partial(
    const int* __restrict__ src, const int* __restrict__ dst,
    const float* __restrict__ h, const unsigned* __restrict__ deg,
    float* __restrict__ partial)
{
    __shared__ float red[AGG_THREADS * HIDDEN];   // 16 KB
    float acc[HIDDEN];
    #pragma unroll
    for (int j = 0; j < HIDDEN; ++j) acc[j] = 0.0f;

    int i = blockIdx.x * blockDim.x + threadIdx.x;
    int stride = gridDim.x * blockDim.x;
    for (int e = i; e < N_EDGES; e += stride) {
        const int s = src[e], d = dst[e];
        const unsigned dg = deg[s];
        const float w = dg ? (1.0f / (float)dg) : 0.0f;
        const float4* hp = (const float4*)(h + (size_t)d * HIDDEN);
        #pragma unroll
        for (int q = 0; q < 4; ++q) {
            const float4 hv = hp[q];
            acc[q * 4 + 0] += hv.x * w;
            acc[q * 4 + 1] += hv.y * w;
            acc[q * 4 + 2] += hv.z * w;
            acc[q * 4 + 3] += hv.w * w;
        }
    }
    #pragma unroll
    for (int j = 0; j < HIDDEN; ++j) red[threadIdx.x * HIDDEN + j] = acc[j];
    __syncthreads();
    if (threadIdx.x < HIDDEN) {
        float s = 0.0f;
        for (int t = 0; t < AGG_THREADS; ++t) s += red[t * HIDDEN + threadIdx.x];
        partial[blockIdx.x * HIDDEN + threadIdx.x] = s;
    }
}

// ---------------- final reduce + classifier head ----------------
__global__ void final_kernel(const float* __restrict__ partial,
                             const float* __restrict__ Wc,
                             const float* __restrict__ bc,
                             float* __restrict__ out)
{
    __shared__ float aggs[HIDDEN];
    if (threadIdx.x < HIDDEN) {
        float s = 0.0f;
        for (int b = 0; b < AGG_BLOCKS; ++b) s += partial[b * HIDDEN + threadIdx.x];
        aggs[threadIdx.x] = s / (float)N_NODES;
    }
    __syncthreads();
    if (threadIdx.x == 0) {
        float z = bc[0];
        #pragma unroll
        for (int j = 0; j < HIDDEN; ++j) z += aggs[j] * Wc[j];
        out[0] = 1.0f / (1.0f + expf(-z));
    }
}

extern "C" void kernel_launch(void* const* d_in, const int* in_sizes, int n_in,
                              void* d_out, int out_size, void* d_ws, size_t ws_size,
                              hipStream_t stream) {
    const float* x    = (const float*)d_in[0];
    const int*   esrc = (const int*)d_in[1];
    const int*   edst = (const int*)d_in[2];
    const float* W1   = (const float*)d_in[3];
    const float* b1   = (const float*)d_in[4];
    const float* W2   = (const float*)d_in[5];
    const float* b2   = (const float*)d_in[6];
    const float* W3   = (const float*)d_in[7];
    const float* b3   = (const float*)d_in[8];
    const float* Wc   = (const float*)d_in[9];
    const float* bc   = (const float*)d_in[10];
    float* out = (float*)d_out;

    char* ws = (char*)d_ws;
    float*    h       = (float*)ws;                                   // 6.4 MB
    unsigned* deg     = (unsigned*)(ws + sizeof(float) * (size_t)N_NODES * HIDDEN);
    float*    partial = (float*)(ws + sizeof(float) * (size_t)N_NODES * HIDDEN
                                    + sizeof(unsigned) * (size_t)N_NODES);

    hipLaunchKernelGGL(mlp_kernel, dim3(MLP_BLOCKS), dim3(256), 0, stream,
                       x, W1, b1, W2, b2, W3, b3, h);
    hipLaunchKernelGGL(zero_deg, dim3((N_NODES + 255) / 256), dim3(256), 0, stream, deg);
    hipLaunchKernelGGL(count_deg, dim3(512), dim3(256), 0, stream, esrc, deg);
    hipLaunchKernelGGL(agg_partial, dim3(AGG_BLOCKS), dim3(AGG_THREADS), 0, stream,
                       esrc, edst, h, deg, partial);
    hipLaunchKernelGGL(final_kernel, dim3(1), dim3(32), 0, stream, partial, Wc, bc, out);
}